// Encoder_35416300323168
// MI455X (gfx1250) — compile-verified
//
#include <hip/hip_runtime.h>

typedef __attribute__((ext_vector_type(2))) float v2f;
typedef __attribute__((ext_vector_type(8))) float v8f;
typedef __attribute__((ext_vector_type(4))) unsigned int v4u;
typedef __attribute__((ext_vector_type(8))) int v8i;
typedef __attribute__((ext_vector_type(4))) int v4i;

#define B_ 128
#define T_ 512
#define D_ 300
#define H_ 128
#define G_ 384   // 3*H
#define V_VOCAB 32000

__device__ __forceinline__ float sigmoidf_(float x) {
    return 1.0f / (1.0f + __expf(-x));
}

// ---------------------------------------------------------------------------
// Repack kernel: interleave row pairs of a (2*nPairs, 384) matrix so a WMMA
// B-fragment (rows k,k+1 at one column) is a single contiguous b64 load:
//   dst[(p*384 + col)*2 + r] = src[(2p + r)*384 + col]
// ---------------------------------------------------------------------------
__global__ __launch_bounds__(256)
void pair_rows(const float* __restrict__ src, float* __restrict__ dst, int nPairs)
{
    const int e = blockIdx.x * 256 + threadIdx.x;
    if (e >= nPairs * G_) return;
    const int p   = e / G_;
    const int col = e - p * G_;
    dst[((size_t)p * G_ + col) * 2 + 0] = src[(size_t)(2 * p)     * G_ + col];
    dst[((size_t)p * G_ + col) * 2 + 1] = src[(size_t)(2 * p + 1) * G_ + col];
}

// ---------------------------------------------------------------------------
// Kernel 1: xp[dir][b][t][g] = sum_d emb[idx[b,t], d] * W[d, g] + b_in[g]
// grid = (M/16, 2 dirs), block = 128 (4 waves). The 16 gathered embedding
// rows are DMA'd into LDS by the Tensor Data Mover in gather mode (16-bit
// row indices, tile 16 x 300 fp32), then fp32 WMMA 16x16x4 over K=300.
// ---------------------------------------------------------------------------
__global__ __launch_bounds__(128)
void gru_input_proj(const int* __restrict__ idx,
                    const float* __restrict__ emb,
                    const float* __restrict__ Wt_f, const float* __restrict__ b_f,
                    const float* __restrict__ Wt_b, const float* __restrict__ b_b,
                    float* __restrict__ xp_f, float* __restrict__ xp_b)
{
    __shared__ float As[16 * D_];   // only shared object -> LDS offset 0

    const int dir     = blockIdx.y;
    const float* Wt   = dir ? Wt_b : Wt_f;
    const float* bias = dir ? b_b : b_f;   // row 0 = input bias
    float* xp         = dir ? xp_b : xp_f;

    const int tid   = threadIdx.x;
    const int mBase = blockIdx.x * 16;

    // ---- TDM gather: 16 embedding rows -> LDS (issued by wave 0 only) ----
    if (tid < 32) {
        // Uniform loads of the 16 row indices; force into SGPRs.
        int ri[16];
        #pragma unroll
        for (int r = 0; r < 16; ++r)
            ri[r] = __builtin_amdgcn_readfirstlane(idx[mBase + r]);

        const unsigned long long ga = (unsigned long long)(size_t)emb;

        v4u g0;
        g0.x = 0x80000001u;                      // count=1 | gather_mode | 16-bit idx
        g0.y = 0u;                               // lds_addr = offset of As = 0
        g0.z = (unsigned)(ga & 0xFFFFFFFFu);     // global_addr[31:0]
        g0.w = (unsigned)((ga >> 32) & 0x01FFFFFFu) | (2u << 30); // addr hi | type=2

        v8i g1;
        g1[0] = (2 << 16);                       // data_size = 4 bytes
        g1[1] = (int)((unsigned)D_ << 16);       // tensor_dim0[15:0] (=300)
        g1[2] = (int)(((unsigned)V_VOCAB & 0xFFFFu) << 16); // tensor_dim1[15:0]
        g1[3] = (int)((unsigned)D_ << 16);       // tile_dim0 = 300
        g1[4] = 16;                              // tile_dim1 = #valid indices
        g1[5] = D_;                              // tensor_dim0_stride = 300
        g1[6] = 0;
        g1[7] = 0;

        v4i g2, g3;
        #pragma unroll
        for (int r = 0; r < 4; ++r)
            g2[r] = (ri[2 * r] & 0xFFFF) | (ri[2 * r + 1] << 16);
        #pragma unroll
        for (int r = 0; r < 4; ++r)
            g3[r] = (ri[8 + 2 * r] & 0xFFFF) | (ri[9 + 2 * r] << 16);

#if __clang_major__ >= 23
        v8i gpad = {};
        __builtin_amdgcn_tensor_load_to_lds(g0, g1, g2, g3, gpad, 0);
#else
        __builtin_amdgcn_tensor_load_to_lds(g0, g1, g2, g3, 0);
#endif
        __builtin_amdgcn_s_wait_tensorcnt(0);
    }
    __syncthreads();

    // The TDM wrote As behind the compiler's back (descriptor carries the LDS
    // address as an integer). Escape As into an asm with a memory clobber so
    // its loads are not folded to undef / eliminated.
    {
        float* asp = &As[0];
        asm volatile("" : "+r"(asp) : : "memory");
    }

    const int lane  = tid & 31;
    const int half  = lane >> 4;   // 0: K rows 0,1   1: K rows 2,3
    const int l16   = lane & 15;
    const int nBase = (tid >> 5) * 96;   // 96 cols per wave (6 tiles)

    v8f acc[6] = {};

    for (int kk = 0; kk < D_; kk += 4) {
        const int koff = kk + 2 * half;
        const int p    = (kk >> 1) + half;          // paired-row index in Wt
        const v2f af = *(const v2f*)&As[l16 * D_ + koff];
        #pragma unroll
        for (int j = 0; j < 6; ++j) {
            const int col = nBase + j * 16 + l16;
            const v2f bf = *(const v2f*)&Wt[((size_t)p * G_ + col) * 2];
            acc[j] = __builtin_amdgcn_wmma_f32_16x16x4_f32(
                false, af, false, bf, (short)0, acc[j], false, false);
        }
    }

    // Epilogue: bias add + store. C layout: VGPR v, lanes<16 -> M=v, >=16 -> M=v+8.
    #pragma unroll
    for (int j = 0; j < 6; ++j) {
        const int col = nBase + j * 16 + l16;
        const float bi = bias[col];
        #pragma unroll
        for (int v = 0; v < 8; ++v) {
            const int row = v + 8 * half;
            xp[(size_t)(mBase + row) * G_ + col] = acc[j][v] + bi;
        }
    }
}

// ---------------------------------------------------------------------------
// Kernel 2: sequential GRU scan. grid = 16 blocks (8 batch-groups x 2 dirs),
// block = 256 (8 waves). Each block owns 16 batch rows; h (16x128) stays in
// LDS across all 512 steps. Per step: rec = h @ U + b_r via fp32 WMMA with
// paired-row Ut (b64 B-fragment loads), then fused gates + output store.
// Backward direction iterates t = T-1..0 (no physical reversal).
// ---------------------------------------------------------------------------
__global__ __launch_bounds__(256)
void gru_scan(const float* __restrict__ state_f, const float* __restrict__ state_b,
              const float* __restrict__ Ut_f, const float* __restrict__ b_f,
              const float* __restrict__ Ut_b, const float* __restrict__ b_b,
              const float* __restrict__ xp_f, const float* __restrict__ xp_b,
              float* __restrict__ out)
{
    __shared__ float hS[16 * H_];    // 8 KB  : hidden state
    __shared__ float recS[16 * G_];  // 24 KB : rec = h @ U + b_r

    const int dir     = blockIdx.x >> 3;
    const int bBase   = (blockIdx.x & 7) * 16;
    const float* Ut   = dir ? Ut_b : Ut_f;
    const float* bias = (dir ? b_b : b_f) + G_;   // row 1 = recurrent bias
    const float* xp   = dir ? xp_b : xp_f;
    const float* h0   = dir ? state_b : state_f;

    const int tid   = threadIdx.x;
    const int lane  = tid & 31;
    const int half  = lane >> 4;
    const int l16   = lane & 15;
    const int nBase = (tid >> 5) * 48;   // 48 cols per wave (3 tiles)

    for (int e = tid; e < 16 * H_; e += 256)
        hS[e] = h0[(size_t)(bBase + (e >> 7)) * H_ + (e & 127)];
    __syncthreads();

    float* hOut = out + (size_t)B_ * T_ * (2 * H_) + (size_t)dir * B_ * H_;

    for (int s = 0; s < T_; ++s) {
        const int t = dir ? (T_ - 1 - s) : s;

        // Prefetch next step's xp slice (16 rows x 384 floats, 96B/thread).
        if (s + 1 < T_) {
            const int tn  = dir ? (t - 1) : (t + 1);
            const int row = tid >> 4;
            const int off = (tid & 15) * 24;
            __builtin_prefetch(&xp[((size_t)(bBase + row) * T_ + tn) * G_ + off], 0, 1);
        }

        // rec = h @ U  (M=16, N=48 per wave, K=128)
        v8f acc[3] = {};
        for (int kk = 0; kk < H_; kk += 4) {
            const int koff = kk + 2 * half;
            const int p    = (kk >> 1) + half;
            const v2f af = *(const v2f*)&hS[l16 * H_ + koff];
            #pragma unroll
            for (int j = 0; j < 3; ++j) {
                const int col = nBase + j * 16 + l16;
                const v2f bf = *(const v2f*)&Ut[((size_t)p * G_ + col) * 2];
                acc[j] = __builtin_amdgcn_wmma_f32_16x16x4_f32(
                    false, af, false, bf, (short)0, acc[j], false, false);
            }
        }
        #pragma unroll
        for (int j = 0; j < 3; ++j) {
            const int col = nBase + j * 16 + l16;
            const float br = bias[col];
            #pragma unroll
            for (int v = 0; v < 8; ++v) {
                const int row = v + 8 * half;
                recS[row * G_ + col] = acc[j][v] + br;
            }
        }
        __syncthreads();

        // Fused gates: 16 rows x 128 cols, 8 elements per thread.
        for (int e = tid; e < 16 * H_; e += 256) {
            const int row = e >> 7;
            const int n   = e & 127;
            const size_t xbase = ((size_t)(bBase + row) * T_ + t) * G_;
            const float xz = xp[xbase + n];
            const float xr = xp[xbase + H_ + n];
            const float xh = xp[xbase + 2 * H_ + n];
            const float rz = recS[row * G_ + n];
            const float rr = recS[row * G_ + H_ + n];
            const float rh = recS[row * G_ + 2 * H_ + n];
            const float z  = sigmoidf_(xz + rz);
            const float r  = sigmoidf_(xr + rr);
            const float hh = tanhf(xh + r * rh);
            const float hp = hS[e];
            const float hn = z * hp + (1.0f - z) * hh;
            hS[e] = hn;
            out[((size_t)(bBase + row) * T_ + t) * (2 * H_) + dir * H_ + n] = hn;
        }
        __syncthreads();
    }

    for (int e = tid; e < 16 * H_; e += 256)
        hOut[(size_t)(bBase + (e >> 7)) * H_ + (e & 127)] = hS[e];
}

// ---------------------------------------------------------------------------
extern "C" void kernel_launch(void* const* d_in, const int* in_sizes, int n_in,
                              void* d_out, int out_size, void* d_ws, size_t ws_size,
                              hipStream_t stream)
{
    const int*   idx     = (const int*)  d_in[0];
    const float* state_f = (const float*)d_in[1];
    const float* state_b = (const float*)d_in[2];
    const float* emb     = (const float*)d_in[3];
    const float* W_f     = (const float*)d_in[4];
    const float* U_f     = (const float*)d_in[5];
    const float* b_f     = (const float*)d_in[6];
    const float* W_b     = (const float*)d_in[7];
    const float* U_b     = (const float*)d_in[8];
    const float* b_b     = (const float*)d_in[9];

    float* out  = (float*)d_out;

    // Workspace layout (fp32): xp_f | xp_b | Wt_f | Wt_b | Ut_f | Ut_b
    float* xp_f = (float*)d_ws;
    float* xp_b = xp_f + (size_t)B_ * T_ * G_;
    float* Wt_f = xp_b + (size_t)B_ * T_ * G_;
    float* Wt_b = Wt_f + (size_t)(D_ / 2) * G_ * 2;
    float* Ut_f = Wt_b + (size_t)(D_ / 2) * G_ * 2;
    float* Ut_b = Ut_f + (size_t)(H_ / 2) * G_ * 2;

    const int wElems = (D_ / 2) * G_;   // 150 * 384
    const int uElems = (H_ / 2) * G_;   // 64 * 384
    pair_rows<<<(wElems + 255) / 256, 256, 0, stream>>>(W_f, Wt_f, D_ / 2);
    pair_rows<<<(wElems + 255) / 256, 256, 0, stream>>>(W_b, Wt_b, D_ / 2);
    pair_rows<<<(uElems + 255) / 256, 256, 0, stream>>>(U_f, Ut_f, H_ / 2);
    pair_rows<<<(uElems + 255) / 256, 256, 0, stream>>>(U_b, Ut_b, H_ / 2);

    dim3 g1((B_ * T_) / 16, 2);
    gru_input_proj<<<g1, 128, 0, stream>>>(idx, emb, Wt_f, b_f, Wt_b, b_b, xp_f, xp_b);
    gru_scan<<<16, 256, 0, stream>>>(state_f, state_b, Ut_f, b_f, Ut_b, b_b,
                                     xp_f, xp_b, out);
}